// Deepmd_radius_62328565399853
// MI455X (gfx1250) — compile-verified
//
#include <hip/hip_runtime.h>

// Problem constants (match reference setup_inputs / globals)
#define BB   8
#define NN   8192
#define KNEI 256
#define NRAD 300
#define RCUT 6.0f
#define PI_OVER_RC 0.52359877559829887308f  // pi / 6.0

// One wave32 per row; logical sort index idx = lane*8 + r  (lane = high bits).
// Bitonic strides j<8  -> intra-lane register min/max swaps (pure VALU)
//                 j>=8 -> __shfl_xor with lane mask j>>3   (DS swizzle)
__global__ __launch_bounds__(256)
void deepmd_radius_sort_kernel(const float* __restrict__ pos,     // [B,N,3]
                               const float* __restrict__ cell,    // [B,3,3]
                               const int*   __restrict__ neigh,   // [B,N,Knei]
                               const float* __restrict__ mask,    // [B,N,Knei]
                               const float* __restrict__ offs,    // [B,N,Knei,3]
                               float*       __restrict__ out)     // [B,N,NRAD]
{
    const int lane = threadIdx.x & 31;
    const int wave = threadIdx.x >> 5;
    const int row  = blockIdx.x * 8 + wave;          // 0 .. B*N-1 (65536)
    const int b    = row >> 13;                      // N = 8192 = 2^13

    // ---- wave-uniform data: 3x3 cell and central atom position ----
    const float* cb = cell + b * 9;
    const float c00 = cb[0], c01 = cb[1], c02 = cb[2];
    const float c10 = cb[3], c11 = cb[4], c12 = cb[5];
    const float c20 = cb[6], c21 = cb[7], c22 = cb[8];

    const float* pc = pos + (size_t)row * 3;
    const float px = pc[0], py = pc[1], pz = pc[2];

    const size_t rk   = (size_t)row * KNEI;
    const int*   nrow = neigh + rk;
    const float* mrow = mask  + rk;
    const float* orow = offs  + rk * 3;
    const float* posb = pos + (size_t)b * NN * 3;

    const int kbase = lane * 8;                      // 8 consecutive neighbors/lane

    // ---- vectorized neighbor-index load (2 x b128), prefetch gather targets ----
    int ji[8];
    {
        const int4* np = (const int4*)(nrow + kbase);
        int4 j0 = np[0], j1 = np[1];
        ji[0] = j0.x; ji[1] = j0.y; ji[2] = j0.z; ji[3] = j0.w;
        ji[4] = j1.x; ji[5] = j1.y; ji[6] = j1.z; ji[7] = j1.w;
    }
#pragma unroll
    for (int r = 0; r < 8; ++r)
        __builtin_prefetch(posb + (size_t)ji[r] * 3, 0, 3);  // global_prefetch_b8 (near)

    // ---- vectorized mask (2 x b128) and offsets (6 x b128) ----
    float mm[8];
    {
        const float4* mp = (const float4*)(mrow + kbase);
        float4 m0 = mp[0], m1 = mp[1];
        mm[0] = m0.x; mm[1] = m0.y; mm[2] = m0.z; mm[3] = m0.w;
        mm[4] = m1.x; mm[5] = m1.y; mm[6] = m1.z; mm[7] = m1.w;
    }
    float oa[24];
    {
        const float4* op = (const float4*)(orow + (size_t)kbase * 3);
#pragma unroll
        for (int i = 0; i < 6; ++i)
            *(float4*)&oa[4 * i] = op[i];
    }

    // ---- per-neighbor cutoff value ----
    float v[8];
#pragma unroll
    for (int r = 0; r < 8; ++r) {
        const float m  = mm[r];
        const float o0 = oa[3 * r + 0];
        const float o1 = oa[3 * r + 1];
        const float o2 = oa[3 * r + 2];
        const float* pj = posb + (size_t)ji[r] * 3;
        const float dx = pj[0] - px + o0 * c00 + o1 * c10 + o2 * c20;
        const float dy = pj[1] - py + o0 * c01 + o1 * c11 + o2 * c21;
        const float dz = pj[2] - pz + o0 * c02 + o1 * c12 + o2 * c22;
        const float d2 = fmaxf(dx * dx + dy * dy + dz * dz, 1e-12f);
        const float d  = sqrtf(d2);                        // v_sqrt_f32
        float cu = 0.5f * (__cosf(d * PI_OVER_RC) + 1.0f); // v_cos_f32
        v[r] = (d < RCUT && m != 0.0f) ? cu : 0.0f;
    }

    // ---- 256-element descending bitonic sort, idx = lane*8 + r ----
#pragma unroll
    for (int k = 2; k <= 256; k <<= 1) {
#pragma unroll
        for (int j = k >> 1; j > 0; j >>= 1) {
            if (j < 8) {
                // intra-lane: partner differs in register bits
#pragma unroll
                for (int r = 0; r < 8; ++r) {
                    const int r2 = r ^ j;
                    if (r < r2) {
                        const bool down = ((((lane << 3) | r) & k) == 0);
                        const float a = v[r], c = v[r2];
                        const float mx = fmaxf(a, c), mn = fminf(a, c);
                        v[r]  = down ? mx : mn;
                        v[r2] = down ? mn : mx;
                    }
                }
            } else {
                // cross-lane: partner differs in lane bits (mask = j/8)
                const int lmask = j >> 3;
#pragma unroll
                for (int r = 0; r < 8; ++r) {
                    const float other = __shfl_xor(v[r], lmask, 32);
                    const bool down   = (((lane << 3) & k) == 0); // k>=16 here: r bits drop out
                    const bool upper  = (lane & lmask) != 0;
                    const bool keepmx = (down != upper);
                    const float mx = fmaxf(v[r], other), mn = fminf(v[r], other);
                    v[r] = keepmx ? mx : mn;
                }
            }
        }
    }

    // ---- write sorted row (2 x b128) + zero padding to NRAD=300 (11 x b128) ----
    float* orow_out = out + (size_t)row * NRAD;
    {
        float4* sp = (float4*)(orow_out + kbase);
        float4 s0, s1;
        s0.x = v[0]; s0.y = v[1]; s0.z = v[2]; s0.w = v[3];
        s1.x = v[4]; s1.y = v[5]; s1.z = v[6]; s1.w = v[7];
        sp[0] = s0;
        sp[1] = s1;
    }
    if (lane < 11) {
        float4 z; z.x = 0.0f; z.y = 0.0f; z.z = 0.0f; z.w = 0.0f;
        ((float4*)(orow_out + KNEI))[lane] = z;   // entries [256, 300)
    }
}

extern "C" void kernel_launch(void* const* d_in, const int* in_sizes, int n_in,
                              void* d_out, int out_size, void* d_ws, size_t ws_size,
                              hipStream_t stream) {
    const float* positions = (const float*)d_in[0];   // [8,8192,3] f32
    const float* cell      = (const float*)d_in[1];   // [8,3,3]    f32
    const int*   neighbors = (const int*)  d_in[2];   // [8,8192,256] i32
    const float* mask      = (const float*)d_in[3];   // [8,8192,256] f32
    const float* offsets   = (const float*)d_in[4];   // [8,8192,256,3] f32
    // d_in[5] = atomic_numbers (unused by reference output)
    float* out = (float*)d_out;                       // [8,8192,300] f32

    const int rows = BB * NN;                         // 65536
    dim3 grid(rows / 8), block(256);                  // 8 waves (rows) per block
    hipLaunchKernelGGL(deepmd_radius_sort_kernel, grid, block, 0, stream,
                       positions, cell, neighbors, mask, offsets, out);
}